// PSELoss_44538810860006
// MI455X (gfx1250) — compile-verified
//
#include <hip/hip_runtime.h>

#define NPIX 409600
#define N4   102400         // NPIX / 4 (float4 elements per image plane)
#define HB   40             // histogram blocks per image (N4/HB = 2560)
#define DC   64             // dice chunks per image      (N4/DC = 1600)
#define EPSV 1e-6f

typedef __attribute__((ext_vector_type(2))) float v2f;
typedef __attribute__((ext_vector_type(8))) float v8f;

// ---- workspace layout (bytes) ----
#define OFF_STATS 0u                                   // pos[16], negtot[16] (u32)
#define OFF_HIST0 256u                                 // 16 * 4096 u32
#define OFF_HIST1 (OFF_HIST0 + 16u*4096u*4u)           // 16 * 4096 u32
#define OFF_HIST2 (OFF_HIST1 + 16u*4096u*4u)           // 16 * 256 u32
#define OFF_STATE (OFF_HIST2 + 16u*256u*4u)            // 16 * uint4 {prefix, rank, use, thr}
#define OFF_PART  (OFF_STATE + 16u*16u)                // 336 stats * 64 chunks (f32)
#define OFF_SUMS  (OFF_PART + 336u*64u*4u)             // 336 f32
#define OFF_END   (OFF_SUMS + 336u*4u)

// monotone key: ascending uint order == ascending float order
__device__ __forceinline__ unsigned keyOf(float x) {
  unsigned u = __float_as_uint(x);
  return (u & 0x80000000u) ? ~u : (u | 0x80000000u);
}
__device__ __forceinline__ float fromKey(unsigned k) {
  unsigned u = (k & 0x80000000u) ? (k ^ 0x80000000u) : ~k;
  return __uint_as_float(u);
}
__device__ __forceinline__ unsigned waveSumU(unsigned v) {
  for (int o = 16; o > 0; o >>= 1) v += __shfl_down(v, o, 32);
  return v;
}
__device__ __forceinline__ float waveSumF(float v) {
  for (int o = 16; o > 0; o >>= 1) v += __shfl_down(v, o, 32);
  return v;
}
__device__ __forceinline__ float sigmoidf_fast(float x) {
  return 1.f / (1.f + __expf(-x));
}

// ---------------- Pass 0: counts + top-12-bit histogram of negative preds ----------------
__global__ void k_pass0(const float4* __restrict__ pred, const float4* __restrict__ gt,
                        const float4* __restrict__ tm, unsigned* __restrict__ stats,
                        unsigned* __restrict__ hist0) {
  __shared__ unsigned h[4096];
  __shared__ unsigned shPos, shNeg;
  const int b = blockIdx.x, chunk = blockIdx.y, tid = threadIdx.x;
  for (int j = tid; j < 4096; j += 256) h[j] = 0u;
  if (tid == 0) { shPos = 0u; shNeg = 0u; }
  __syncthreads();

  const float4* pt = pred + ((size_t)b * 7 + 6) * N4;
  const float4* g4 = gt + (size_t)b * N4;
  const float4* t4 = tm + (size_t)b * N4;
  unsigned pos = 0, neg = 0;
  const int beg = chunk * (N4 / HB), end = beg + (N4 / HB);
  for (int i = beg + tid; i < end; i += 256) {
    float4 p = pt[i], g = g4[i], t = t4[i];
    float pv[4] = {p.x, p.y, p.z, p.w};
    float gv[4] = {g.x, g.y, g.z, g.w};
    float tv[4] = {t.x, t.y, t.z, t.w};
#pragma unroll
    for (int c = 0; c < 4; ++c) {
      if (gv[c] > 0.5f && tv[c] > 0.5f) pos++;
      if (gv[c] <= 0.5f) { neg++; atomicAdd(&h[keyOf(pv[c]) >> 20], 1u); }
    }
  }
  unsigned pw = waveSumU(pos), nw = waveSumU(neg);
  if ((tid & 31) == 0) { atomicAdd(&shPos, pw); atomicAdd(&shNeg, nw); }
  __syncthreads();
  if (tid == 0) { atomicAdd(&stats[b], shPos); atomicAdd(&stats[16 + b], shNeg); }
  for (int j = tid; j < 4096; j += 256)
    if (h[j]) atomicAdd(&hist0[b * 4096 + j], h[j]);
}

// ---------------- Select level 0 ----------------
__global__ void k_sel0(const unsigned* __restrict__ stats, const unsigned* __restrict__ hist0,
                       uint4* __restrict__ state) {
  const int b = blockIdx.x;
  unsigned pos = stats[b], negt = stats[16 + b];
  unsigned nn3 = pos * 3u;
  unsigned neg_num = nn3 < negt ? nn3 : negt;
  unsigned use = (pos > 0u && neg_num > 0u) ? 1u : 0u;
  uint4 st; st.x = 0u; st.y = 0u; st.z = use; st.w = 0u;
  if (use) {
    unsigned r = negt - neg_num;   // ascending 0-based rank of k-th largest
    unsigned bin = 4095u;
    for (int j = 0; j < 4096; ++j) {
      unsigned c = hist0[b * 4096 + j];
      if (r < c) { bin = (unsigned)j; break; }
      r -= c;
    }
    st.x = bin << 20; st.y = r;
  }
  state[b] = st;
}

// ---------------- Pass 1: middle-12-bit histogram within selected bin ----------------
__global__ void k_pass1(const float4* __restrict__ pred, const float4* __restrict__ gt,
                        const uint4* __restrict__ state, unsigned* __restrict__ hist1) {
  const int b = blockIdx.x, chunk = blockIdx.y, tid = threadIdx.x;
  uint4 st = state[b];
  if (st.z == 0u) return;
  const unsigned pref = st.x >> 20;
  __shared__ unsigned h[4096];
  for (int j = tid; j < 4096; j += 256) h[j] = 0u;
  __syncthreads();
  const float4* pt = pred + ((size_t)b * 7 + 6) * N4;
  const float4* g4 = gt + (size_t)b * N4;
  const int beg = chunk * (N4 / HB), end = beg + (N4 / HB);
  for (int i = beg + tid; i < end; i += 256) {
    float4 p = pt[i], g = g4[i];
    float pv[4] = {p.x, p.y, p.z, p.w};
    float gv[4] = {g.x, g.y, g.z, g.w};
#pragma unroll
    for (int c = 0; c < 4; ++c) {
      if (gv[c] <= 0.5f) {
        unsigned k = keyOf(pv[c]);
        if ((k >> 20) == pref) atomicAdd(&h[(k >> 8) & 0xFFFu], 1u);
      }
    }
  }
  __syncthreads();
  for (int j = tid; j < 4096; j += 256)
    if (h[j]) atomicAdd(&hist1[b * 4096 + j], h[j]);
}

__global__ void k_sel1(const unsigned* __restrict__ hist1, uint4* __restrict__ state) {
  const int b = blockIdx.x;
  uint4 st = state[b];
  if (st.z == 0u) return;
  unsigned r = st.y, bin = 4095u;
  for (int j = 0; j < 4096; ++j) {
    unsigned c = hist1[b * 4096 + j];
    if (r < c) { bin = (unsigned)j; break; }
    r -= c;
  }
  st.x |= bin << 8; st.y = r;
  state[b] = st;
}

// ---------------- Pass 2: low-8-bit histogram within 24-bit prefix ----------------
__global__ void k_pass2(const float4* __restrict__ pred, const float4* __restrict__ gt,
                        const uint4* __restrict__ state, unsigned* __restrict__ hist2) {
  const int b = blockIdx.x, chunk = blockIdx.y, tid = threadIdx.x;
  uint4 st = state[b];
  if (st.z == 0u) return;
  const unsigned pref = st.x >> 8;
  __shared__ unsigned h[256];
  if (tid < 256) h[tid] = 0u;
  __syncthreads();
  const float4* pt = pred + ((size_t)b * 7 + 6) * N4;
  const float4* g4 = gt + (size_t)b * N4;
  const int beg = chunk * (N4 / HB), end = beg + (N4 / HB);
  for (int i = beg + tid; i < end; i += 256) {
    float4 p = pt[i], g = g4[i];
    float pv[4] = {p.x, p.y, p.z, p.w};
    float gv[4] = {g.x, g.y, g.z, g.w};
#pragma unroll
    for (int c = 0; c < 4; ++c) {
      if (gv[c] <= 0.5f) {
        unsigned k = keyOf(pv[c]);
        if ((k >> 8) == pref) atomicAdd(&h[k & 0xFFu], 1u);
      }
    }
  }
  __syncthreads();
  if (tid < 256 && h[tid]) atomicAdd(&hist2[b * 256 + tid], h[tid]);
}

__global__ void k_sel2(const unsigned* __restrict__ hist2, uint4* __restrict__ state) {
  const int b = blockIdx.x;
  uint4 st = state[b];
  if (st.z == 0u) return;
  unsigned r = st.y, bin = 255u;
  for (int j = 0; j < 256; ++j) {
    unsigned c = hist2[b * 256 + j];
    if (r < c) { bin = (unsigned)j; break; }
    r -= c;
  }
  st.w = __float_as_uint(fromKey(st.x | bin));   // exact threshold value
  state[b] = st;
}

// ---------------- Fused dice pass: 21 stats per (image, chunk) ----------------
__global__ void k_dice(const float4* __restrict__ pred, const float4* __restrict__ gtT,
                       const float4* __restrict__ gtK, const float4* __restrict__ tm,
                       const uint4* __restrict__ state, float* __restrict__ part) {
  const int b = blockIdx.x, chunk = blockIdx.y, tid = threadIdx.x;
  uint4 st = state[b];
  const int use = (int)st.z;
  const float thr = __uint_as_float(st.w);

  const float4* pt = pred + ((size_t)b * 7 + 6) * N4;
  const float4* g4 = gtT + (size_t)b * N4;
  const float4* t4 = tm + (size_t)b * N4;

  float acc[21];
#pragma unroll
  for (int j = 0; j < 21; ++j) acc[j] = 0.f;

  const int beg = chunk * (N4 / DC), end = beg + (N4 / DC);
  for (int i = beg + tid; i < end; i += 256) {
    float4 p = pt[i], g = g4[i], t = t4[i];
    float pv[4] = {p.x, p.y, p.z, p.w};
    float gv[4] = {g.x, g.y, g.z, g.w};
    float tv[4] = {t.x, t.y, t.z, t.w};
    float km[4];
#pragma unroll
    for (int c = 0; c < 4; ++c) {
      float s = sigmoidf_fast(pv[c]);
      // sel mask: binary OHEM mask (m^2 == m) or raw float tm (mask^2 = tm^2)
      float mm = use ? ((((pv[c] >= thr) || (gv[c] > 0.5f)) && (tv[c] > 0.5f)) ? 1.f : 0.f)
                     : tv[c] * tv[c];
      acc[0] += s * gv[c] * mm;        // a_text
      acc[1] += s * s * mm;            // b_text
      acc[2] += gv[c] * gv[c] * mm;    // c_text
      km[c] = (pv[c] > 0.f && tv[c] > 0.5f) ? 1.f : 0.f;  // sigmoid>0.5 <=> pred>0
    }
#pragma unroll
    for (int k = 0; k < 6; ++k) {
      float4 pk = pred[((size_t)b * 7 + k) * N4 + i];
      float4 gk = gtK[((size_t)b * 6 + k) * N4 + i];
      float pkv[4] = {pk.x, pk.y, pk.z, pk.w};
      float gkv[4] = {gk.x, gk.y, gk.z, gk.w};
#pragma unroll
      for (int c = 0; c < 4; ++c) {
        float sk = sigmoidf_fast(pkv[c]);
        acc[3 + 3 * k] += sk * gkv[c] * km[c];
        acc[4 + 3 * k] += sk * sk * km[c];
        acc[5 + 3 * k] += gkv[c] * gkv[c] * km[c];
      }
    }
  }

  // deterministic block reduction: wave32 shuffle, then per-wave partials in LDS
  __shared__ float red[8][21];
  const int wv = tid >> 5, lane = tid & 31;
#pragma unroll
  for (int j = 0; j < 21; ++j) {
    float s = waveSumF(acc[j]);
    if (lane == 0) red[wv][j] = s;
  }
  __syncthreads();
  if (tid < 21) {
    float s = 0.f;
#pragma unroll
    for (int w = 0; w < 8; ++w) s += red[w][tid];
    part[((size_t)tid * 16 + b) * 64 + chunk] = s;   // stat (tid,b), chunk column
  }
}

// ---------------- WMMA cross-block reduction: 336 stats x 64 partials ----------------
// D = A(16x4 partials) x B(4x16 ones) + C ; D[m][n] accumulates the row-sum of A
// A layout (ISA 7.12.2, f32 16x4): lanes 0-15 -> M=0..15, vgpr0=K0, vgpr1=K1;
//                                  lanes 16-31 -> M=0..15, vgpr0=K2, vgpr1=K3.
// B = ones => layout-independent. D[m][0]: m<8 -> c[m]@lane0, m>=8 -> c[m-8]@lane16.
__global__ void k_wmma_reduce(const float* __restrict__ part, float* __restrict__ sums) {
  const int lane = threadIdx.x;          // one wave of 32
  const int m = lane & 15;
  const int kb = (lane >> 4) * 2;        // K base: 0 or 2
  v2f ones; ones.x = 1.f; ones.y = 1.f;
  for (int g = 0; g < 21; ++g) {
    v8f c = {};
    const float* base = part + ((size_t)g * 16 + m) * 64 + kb;
#pragma unroll
    for (int it = 0; it < 16; ++it) {
      v2f a;
      a.x = base[it * 4];
      a.y = base[it * 4 + 1];
      c = __builtin_amdgcn_wmma_f32_16x16x4_f32(false, a, false, ones,
                                                (short)0, c, false, false);
    }
    if (lane == 0) {
#pragma unroll
      for (int r = 0; r < 8; ++r) sums[g * 16 + r] = c[r];
    } else if (lane == 16) {
#pragma unroll
      for (int r = 0; r < 8; ++r) sums[g * 16 + 8 + r] = c[r];
    }
  }
}

// ---------------- Finalize scalar loss ----------------
__global__ void k_final(const float* __restrict__ sums, float* __restrict__ out) {
  if (threadIdx.x == 0 && blockIdx.x == 0) {
    float lt = 0.f;
    for (int b = 0; b < 16; ++b) {
      float a = sums[b], bb = sums[16 + b], cc = sums[32 + b];
      lt += 1.f - 2.f * a / (bb + cc + 2.f * EPSV);
    }
    lt *= (1.f / 16.f);
    float lk = 0.f;
    for (int k = 0; k < 6; ++k)
      for (int b = 0; b < 16; ++b) {
        float a = sums[(3 + 3 * k) * 16 + b];
        float bb = sums[(4 + 3 * k) * 16 + b];
        float cc = sums[(5 + 3 * k) * 16 + b];
        lk += 1.f - 2.f * a / (bb + cc + 2.f * EPSV);
      }
    lk *= (1.f / 96.f);
    out[0] = 0.7f * lt + 0.3f * lk;
  }
}

extern "C" void kernel_launch(void* const* d_in, const int* in_sizes, int n_in,
                              void* d_out, int out_size, void* d_ws, size_t ws_size,
                              hipStream_t stream) {
  (void)in_sizes; (void)n_in; (void)out_size; (void)ws_size;
  const float4* pred = (const float4*)d_in[0];   // [16,7,640,640]
  const float4* gtT  = (const float4*)d_in[1];   // [16,640,640]
  const float4* gtK  = (const float4*)d_in[2];   // [16,6,640,640]
  const float4* tmm  = (const float4*)d_in[3];   // [16,640,640]

  char* ws = (char*)d_ws;
  unsigned* stats = (unsigned*)(ws + OFF_STATS);
  unsigned* hist0 = (unsigned*)(ws + OFF_HIST0);
  unsigned* hist1 = (unsigned*)(ws + OFF_HIST1);
  unsigned* hist2 = (unsigned*)(ws + OFF_HIST2);
  uint4*    state = (uint4*)(ws + OFF_STATE);
  float*    part  = (float*)(ws + OFF_PART);
  float*    sums  = (float*)(ws + OFF_SUMS);

  hipMemsetAsync(d_ws, 0, OFF_END, stream);   // graph-capture-safe re-zero each replay

  k_pass0<<<dim3(16, HB), 256, 0, stream>>>(pred, gtT, tmm, stats, hist0);
  k_sel0<<<16, 1, 0, stream>>>(stats, hist0, state);
  k_pass1<<<dim3(16, HB), 256, 0, stream>>>(pred, gtT, state, hist1);
  k_sel1<<<16, 1, 0, stream>>>(hist1, state);
  k_pass2<<<dim3(16, HB), 256, 0, stream>>>(pred, gtT, state, hist2);
  k_sel2<<<16, 1, 0, stream>>>(hist2, state);
  k_dice<<<dim3(16, DC), 256, 0, stream>>>(pred, gtT, gtK, tmm, state, part);
  k_wmma_reduce<<<1, 32, 0, stream>>>(part, sums);
  k_final<<<1, 1, 0, stream>>>(sums, (float*)d_out);
}